// Decoder_20014547599495
// MI455X (gfx1250) — compile-verified
//
#include <hip/hip_runtime.h>
#include <hip/hip_bf16.h>

// ---------------------------------------------------------------------------
// CDNA5 / gfx1250 fp32 WMMA decoder step (attention + GRU + vocab projection)
// D = A(16x4) * B(4x16) + C(16x16), f32 throughout (matches fp32 reference).
// Lane layout (wave32): ml = lane&15 = row (A) / source row of column (B),
// kg = lane>>4 selects k offset {0,2}; each lane holds a contiguous float2.
// LDS staging uses CDNA5 async loads (global_load_async_to_lds_b128,
// tracked by ASYNCcnt, drained with s_wait_asynccnt).
// ---------------------------------------------------------------------------

typedef __attribute__((ext_vector_type(2))) float v2f;
typedef __attribute__((ext_vector_type(8))) float v8f;

#define VOCAB 32000
#define EMBED 512
#define HID   1024
#define BATCH 64
#define SLEN  512
#define MTOT  (SLEN * BATCH)   // 32768 flattened rows (s*B + b)

static __device__ __forceinline__ v8f wmma_f32(v2f a, v2f b, v8f c) {
  return __builtin_amdgcn_wmma_f32_16x16x4_f32(
      /*neg_a=*/false, a, /*neg_b=*/false, b,
      /*c_mod=*/(short)0, c, /*reuse_a=*/false, /*reuse_b=*/false);
}

static __device__ __forceinline__ float sigmoid_f(float x) {
  return 1.0f / (1.0f + __expf(-x));
}

// Async 16B global -> LDS copy (per-lane addresses), ASYNCcnt-tracked.
static __device__ __forceinline__ void async_ld16(void* lds, const void* g) {
  unsigned off = (unsigned)(uintptr_t)lds;  // low 32 bits = LDS byte offset
  asm volatile("global_load_async_to_lds_b128 %0, %1, off"
               :: "v"(off), "v"(g) : "memory");
}
static __device__ __forceinline__ void wait_async0() {
  asm volatile("s_wait_asynccnt 0" ::: "memory");
}

// ---------------------------------------------------------------------------
// K1: hpre[b][j] = b_attn[j] + sum_k h_last[b][k] * W_attn[j][k]   (W1 half)
// grid(64, 4) block(32): one wave -> one 16x16 tile, K=1024.
// ---------------------------------------------------------------------------
__global__ void k_hpre(const float* __restrict__ h_last,
                       const float* __restrict__ W_attn,
                       const float* __restrict__ b_attn,
                       float* __restrict__ hpre) {
  const int lane = threadIdx.x;
  const int ml = lane & 15, kg = lane >> 4;
  const int nbase = blockIdx.x * 16;
  const int mbase = blockIdx.y * 16;

  const float* Arow = h_last + (mbase + ml) * HID + 2 * kg;
  const float* Brow = W_attn + (nbase + ml) * (2 * HID) + 2 * kg;  // k < HID half

  v8f d = {};
#pragma unroll 4
  for (int k0 = 0; k0 < HID; k0 += 4) {
    v2f a = *(const v2f*)(Arow + k0);
    v2f b = *(const v2f*)(Brow + k0);
    d = wmma_f32(a, b, d);
  }
  const int n = nbase + ml;
  const float bias = b_attn[n];
#pragma unroll
  for (int r = 0; r < 8; ++r) {
    int m = mbase + r + 8 * kg;
    hpre[m * HID + n] = d[r] + bias;
  }
}

// ---------------------------------------------------------------------------
// K2 (dominant): partial[nblk][m] = sum_{j in nblk} tanh(hpre[b][j]
//                                      + enc_m . W2_j) * v[j]
// M=32768, N=1024, K=1024; block tile Bm=128 x Bn=128 (grid 256 x 8):
//   L2 traffic = 1GB(A) + 1GB(B) vs 8GB with Bm=16.
// 8 waves: wave w owns m-tile w, all 8 n-tiles; A-fragment reused 8x.
// A and B tiles staged to LDS with async b128 copies.
// ---------------------------------------------------------------------------
__global__ void k_score(const float* __restrict__ enc,
                        const float* __restrict__ W_attn,
                        const float* __restrict__ v_attn,
                        const float* __restrict__ hpre,
                        float* __restrict__ partial) {
  __shared__ float As[128 * 64];  // [row_local][k_local]   32KB
  __shared__ float Bs[128 * 64];  // [n_local][k_local]     32KB

  const int tid = threadIdx.x;
  const int wave = tid >> 5, lane = tid & 31;
  const int ml = lane & 15, kg = lane >> 4;
  const int mbase = blockIdx.x * 128;
  const int nb0 = blockIdx.y * 128;
  const int mw = mbase + wave * 16;  // this wave's m-tile

  v8f acc[8];
#pragma unroll
  for (int t = 0; t < 8; ++t) acc[t] = (v8f){};

  for (int kc = 0; kc < 16; ++kc) {  // 16 K-chunks of 64
    const int kbase = kc * 64;
    // stage 128x64 of A and of B (2048 float4 each, 8 per thread each)
#pragma unroll
    for (int j = 0; j < 8; ++j) {
      int fidx = tid + j * 256;        // 0..2047 float4
      int row = fidx >> 4;             // 16 float4 per 64-float row
      int c4 = (fidx & 15) * 4;
      async_ld16(As + row * 64 + c4,
                 enc + (size_t)(mbase + row) * HID + kbase + c4);
      async_ld16(Bs + row * 64 + c4,
                 W_attn + (size_t)(nb0 + row) * (2 * HID) + HID + kbase + c4);
    }
    wait_async0();
    __syncthreads();

    const float* Alds = As + (wave * 16 + ml) * 64 + 2 * kg;
    const float* Blds = Bs + ml * 64 + 2 * kg;
#pragma unroll 2
    for (int k0 = 0; k0 < 64; k0 += 4) {
      v2f a = *(const v2f*)(Alds + k0);  // one A fragment, 8 WMMAs
#pragma unroll
      for (int t = 0; t < 8; ++t) {
        v2f b = *(const v2f*)(Blds + t * (16 * 64) + k0);
        acc[t] = wmma_f32(a, b, acc[t]);
      }
    }
    __syncthreads();
  }

  // epilogue: tanh + dot with v_attn over this block's 128 columns
  float part[8];
#pragma unroll
  for (int r = 0; r < 8; ++r) part[r] = 0.0f;
#pragma unroll
  for (int t = 0; t < 8; ++t) {
    const int n = nb0 + t * 16 + ml;
    const float vv = v_attn[n];
#pragma unroll
    for (int r = 0; r < 8; ++r) {
      int m = mw + r + 8 * kg;
      int b = m & (BATCH - 1);
      part[r] += tanhf(acc[t][r] + hpre[b * HID + n]) * vv;
    }
  }
#pragma unroll
  for (int r = 0; r < 8; ++r) {
    part[r] += __shfl_xor(part[r], 8, 32);
    part[r] += __shfl_xor(part[r], 4, 32);
    part[r] += __shfl_xor(part[r], 2, 32);
    part[r] += __shfl_xor(part[r], 1, 32);
  }
  if (ml == 0) {  // lanes 0 (rows +0..7) and 16 (rows +8..15)
#pragma unroll
    for (int r = 0; r < 8; ++r)
      partial[(size_t)blockIdx.y * MTOT + mw + r + 8 * kg] = part[r];
  }
}

// ---------------------------------------------------------------------------
// K3: deterministic partial-sum + softmax over S + write attn_weights (B,S)
// grid(64) block(256), 2 elements/thread.
// ---------------------------------------------------------------------------
__global__ void k_softmax(const float* __restrict__ partial,
                          float* __restrict__ attn,
                          float* __restrict__ attn_w_out) {
  __shared__ float red[256];
  const int b = blockIdx.x, tid = threadIdx.x;
  float x0 = 0.0f, x1 = 0.0f;
#pragma unroll
  for (int p = 0; p < 8; ++p) {
    x0 += partial[(size_t)p * MTOT + tid * BATCH + b];
    x1 += partial[(size_t)p * MTOT + (tid + 256) * BATCH + b];
  }

  red[tid] = fmaxf(x0, x1);
  __syncthreads();
  for (int s = 128; s > 0; s >>= 1) {
    if (tid < s) red[tid] = fmaxf(red[tid], red[tid + s]);
    __syncthreads();
  }
  const float M = red[0];
  __syncthreads();

  float e0 = __expf(x0 - M), e1 = __expf(x1 - M);
  red[tid] = e0 + e1;
  __syncthreads();
  for (int s = 128; s > 0; s >>= 1) {
    if (tid < s) red[tid] += red[tid + s];
    __syncthreads();
  }
  const float inv = 1.0f / red[0];

  float a0 = e0 * inv, a1 = e1 * inv;
  attn[tid * BATCH + b] = a0;
  attn[(tid + 256) * BATCH + b] = a1;
  attn_w_out[b * SLEN + tid] = a0;
  attn_w_out[b * SLEN + tid + 256] = a1;
}

// ---------------------------------------------------------------------------
// K4: context[b][h] = sum_s attn[s][b] * enc[s][b][h]  -> x2[b][HID + h]
// grid(64) block(256), float4 per thread. enc pass should hit L2 (128<192MB).
// ---------------------------------------------------------------------------
__global__ void k_context(const float* __restrict__ enc,
                          const float* __restrict__ attn,
                          float* __restrict__ x2) {
  const int b = blockIdx.x;
  const int c = threadIdx.x * 4;  // 256 threads x 4 floats = 1024
  float4 acc = make_float4(0.f, 0.f, 0.f, 0.f);
#pragma unroll 4
  for (int s = 0; s < SLEN; ++s) {
    float w = attn[s * BATCH + b];
    const float4 e = *(const float4*)(enc + (size_t)(s * BATCH + b) * HID + c);
    acc.x += w * e.x; acc.y += w * e.y; acc.z += w * e.z; acc.w += w * e.w;
  }
  *(float4*)(x2 + b * (2 * HID) + HID + c) = acc;
}

// ---------------------------------------------------------------------------
// K5: gi[b][n] = b_ih + [emb(tok), context] . W_ih[n]    (K = 1536)
//     gh[b][n] = b_hh + h_last . W_hh[n]                 (K = 1024)
// grid(192, 4) block(32): one wave -> one 16x16 tile of both gi and gh.
// ---------------------------------------------------------------------------
__global__ void k_gates_gemm(const int* __restrict__ tok,
                             const float* __restrict__ emb,
                             const float* __restrict__ x2,
                             const float* __restrict__ h_last,
                             const float* __restrict__ W_ih,
                             const float* __restrict__ W_hh,
                             const float* __restrict__ b_ih,
                             const float* __restrict__ b_hh,
                             float* __restrict__ gi,
                             float* __restrict__ gh) {
  const int lane = threadIdx.x;
  const int ml = lane & 15, kg = lane >> 4;
  const int nbase = blockIdx.x * 16;
  const int mbase = blockIdx.y * 16;
  const int n = nbase + ml;

  const int t = tok[mbase + ml];
  const float* Aemb = emb + (size_t)t * EMBED + 2 * kg;
  // A[m][k] for k in [512,1536) = context[m][k-512] = x2[m*2048 + 512 + k]
  const float* Actx = x2 + (mbase + ml) * (2 * HID) + 512 + 2 * kg;
  const float* Bi = W_ih + (size_t)n * (EMBED + HID) + 2 * kg;

  v8f dgi = {};
#pragma unroll 4
  for (int k0 = 0; k0 < EMBED; k0 += 4)
    dgi = wmma_f32(*(const v2f*)(Aemb + k0), *(const v2f*)(Bi + k0), dgi);
#pragma unroll 4
  for (int k0 = EMBED; k0 < EMBED + HID; k0 += 4)
    dgi = wmma_f32(*(const v2f*)(Actx + k0), *(const v2f*)(Bi + k0), dgi);

  const float* Ah = h_last + (mbase + ml) * HID + 2 * kg;
  const float* Bh = W_hh + (size_t)n * HID + 2 * kg;
  v8f dgh = {};
#pragma unroll 4
  for (int k0 = 0; k0 < HID; k0 += 4)
    dgh = wmma_f32(*(const v2f*)(Ah + k0), *(const v2f*)(Bh + k0), dgh);

  const float bi = b_ih[n], bh = b_hh[n];
#pragma unroll
  for (int r = 0; r < 8; ++r) {
    int m = mbase + r + 8 * kg;
    gi[m * (3 * HID) + n] = dgi[r] + bi;
    gh[m * (3 * HID) + n] = dgh[r] + bh;
  }
}

// ---------------------------------------------------------------------------
// K6: GRU gates elementwise -> h_new into x2[:, :HID] and d_out.
// grid(256) block(256): 65536 elements.
// ---------------------------------------------------------------------------
__global__ void k_gru(const float* __restrict__ gi,
                      const float* __restrict__ gh,
                      const float* __restrict__ h_last,
                      float* __restrict__ x2,
                      float* __restrict__ hnew_out) {
  const int idx = blockIdx.x * 256 + threadIdx.x;
  const int b = idx >> 10, j = idx & (HID - 1);
  const int base = b * (3 * HID);
  float r = sigmoid_f(gi[base + j] + gh[base + j]);
  float z = sigmoid_f(gi[base + HID + j] + gh[base + HID + j]);
  float nn = tanhf(gi[base + 2 * HID + j] + r * gh[base + 2 * HID + j]);
  float h = h_last[b * HID + j];
  float hn = (1.0f - z) * nn + z * h;
  x2[b * (2 * HID) + j] = hn;
  hnew_out[b * HID + j] = hn;
}

// ---------------------------------------------------------------------------
// K7: logits[b][n] = b_fc[n] + x2[b] . W_fc[n]   M=64, N=32000, K=2048.
// grid(250, 4) block(256): A tile async-staged in LDS (4 x 32KB chunks),
// 8 waves each own one 16-wide N tile. W_fc (256MB) streamed exactly once:
// HBM-bound floor ~11us @ 23.3TB/s.
// ---------------------------------------------------------------------------
__global__ void k_logits(const float* __restrict__ x2,
                         const float* __restrict__ W_fc,
                         const float* __restrict__ b_fc,
                         float* __restrict__ logits) {
  __shared__ float As[16 * 512];
  const int tid = threadIdx.x;
  const int wave = tid >> 5, lane = tid & 31;
  const int ml = lane & 15, kg = lane >> 4;
  const int mbase = blockIdx.y * 16;
  const int nbase = (blockIdx.x * 8 + wave) * 16;
  const int n = nbase + ml;

  v8f d = {};
  for (int kc = 0; kc < 4; ++kc) {
#pragma unroll
    for (int i = 0; i < 8; ++i) {
      int fidx = tid + i * 256;  // 2048 float4 = 16 x 512 floats
      int row = fidx >> 7;       // 128 float4 per row
      int c4 = (fidx & 127) * 4;
      async_ld16(As + row * 512 + c4,
                 x2 + (size_t)(mbase + row) * (2 * HID) + kc * 512 + c4);
    }
    wait_async0();
    __syncthreads();

    const float* Brow = W_fc + (size_t)n * (2 * HID) + kc * 512 + 2 * kg;
    const float* Alds = As + ml * 512 + 2 * kg;
#pragma unroll 4
    for (int k0 = 0; k0 < 512; k0 += 4) {
      v2f a = *(const v2f*)(Alds + k0);
      v2f b = *(const v2f*)(Brow + k0);
      d = wmma_f32(a, b, d);
    }
    __syncthreads();
  }

  const float bias = b_fc[n];
#pragma unroll
  for (int r = 0; r < 8; ++r) {
    int m = mbase + r + 8 * kg;
    logits[m * VOCAB + n] = d[r] + bias;
  }
}

// ---------------------------------------------------------------------------
extern "C" void kernel_launch(void* const* d_in, const int* in_sizes, int n_in,
                              void* d_out, int out_size, void* d_ws, size_t ws_size,
                              hipStream_t stream) {
  const int*   input_token = (const int*)d_in[0];
  const float* hidden      = (const float*)d_in[1];   // (1,B,H) -> h_last
  const float* enc         = (const float*)d_in[2];   // (S,B,H)
  const float* emb         = (const float*)d_in[3];   // (VOCAB,EMBED)
  const float* W_attn      = (const float*)d_in[4];   // (H, 2H)
  const float* b_attn      = (const float*)d_in[5];
  const float* v_attn      = (const float*)d_in[6];
  const float* W_ih        = (const float*)d_in[7];   // (3H, EMBED+H)
  const float* W_hh        = (const float*)d_in[8];   // (3H, H)
  const float* b_ih        = (const float*)d_in[9];
  const float* b_hh        = (const float*)d_in[10];
  const float* W_fc        = (const float*)d_in[11];  // (VOCAB, 2H)
  const float* b_fc        = (const float*)d_in[12];

  float* out = (float*)d_out;
  float* logits_out = out;                                   // 64*32000
  float* hnew_out   = out + (size_t)BATCH * VOCAB;           // 64*1024
  float* attnw_out  = hnew_out + (size_t)BATCH * HID;        // 64*512

  float* w = (float*)d_ws;
  float* hpre    = w;                  // 65536
  float* partial = hpre + 65536;       // 8 * 32768 = 262144
  float* attn    = partial + 262144;   // 32768
  float* x2      = attn + 32768;       // 131072 ([h_new | context] per row)
  float* gi      = x2 + 131072;        // 196608
  float* gh      = gi + 196608;        // 196608

  k_hpre<<<dim3(HID / 16, BATCH / 16), 32, 0, stream>>>(hidden, W_attn, b_attn, hpre);
  k_score<<<dim3(MTOT / 128, HID / 128), 256, 0, stream>>>(enc, W_attn, v_attn, hpre, partial);
  k_softmax<<<dim3(BATCH), 256, 0, stream>>>(partial, attn, attnw_out);
  k_context<<<dim3(BATCH), 256, 0, stream>>>(enc, attn, x2);
  k_gates_gemm<<<dim3((3 * HID) / 16, BATCH / 16), 32, 0, stream>>>(
      input_token, emb, x2, hidden, W_ih, W_hh, b_ih, b_hh, gi, gh);
  k_gru<<<dim3((BATCH * HID) / 256), 256, 0, stream>>>(gi, gh, hidden, x2, hnew_out);
  k_logits<<<dim3(VOCAB / (8 * 16), BATCH / 16), 256, 0, stream>>>(x2, W_fc, b_fc, logits_out);
}